// GroupedEmbedding_59596966199836
// MI455X (gfx1250) — compile-verified
//
#include <hip/hip_runtime.h>
#include <stdint.h>

// Problem constants (from reference)
#define NUM_TABLES 16
#define ROWS       200000
#define DIM        64          // 64 floats = 256 bytes per row = 16 float4 chunks
#define LOOKUPS    131072      // 2^17
#define TOTAL_ROWS (NUM_TABLES * LOOKUPS)  // 2,097,152 = 2^21

typedef float v4f __attribute__((ext_vector_type(4)));

// ---------------------------------------------------------------------------
// Path B (listed first so the disasm snippet shows it): async LDS pipeline.
// gfx1250 GLOBAL_LOAD_ASYNC_TO_LDS_B128 / GLOBAL_STORE_ASYNC_FROM_LDS_B128,
// tracked by ASYNCcnt.
// A "group" = 2 rows = 512B = 32 lanes x b128.
//   lane: rowInG = lane>>4, chunk = lane&15.
// Each wave stages GPI=8 groups (4KB) in its private LDS slice:
//   issue 8 async gathers -> s_wait_asynccnt 0 -> issue 8 async stores.
// 65536 waves x 4KB in flight ≈ 256MB >> BW*latency product.
// ---------------------------------------------------------------------------
#define GPI 8  // groups per wave

__global__ __launch_bounds__(256) void gather_async(
    const int* __restrict__ idxs, const float* __restrict__ w,
    float* __restrict__ out, unsigned groupBase, unsigned nGroups)
{
    __shared__ char lds[8 /*waves*/ * GPI * 512];   // 32 KB / block

    unsigned lane      = threadIdx.x & 31u;
    unsigned waveInBlk = threadIdx.x >> 5;
    unsigned wave      = blockIdx.x * 8u + waveInBlk;
    unsigned g0loc     = wave * GPI;
    if (g0loc >= nGroups) return;                   // launch math exact: full GPI otherwise
    unsigned g0        = groupBase + g0loc;

    unsigned rowInG = lane >> 4;                    // 0..1
    unsigned chunk  = lane & 15u;                   // 0..15

    // Wave-relative LDS byte address (generic pointer low 32 bits == LDS addr).
    uint32_t ldsBase = (uint32_t)(uintptr_t)(lds + (size_t)waveInBlk * (GPI * 512));
    uint64_t wB   = (uint64_t)(uintptr_t)w;
    uint64_t outB = (uint64_t)(uintptr_t)out;

    // ---- issue GPI async gathers into LDS ----
    #pragma unroll
    for (int k = 0; k < GPI; ++k) {
        unsigned g     = g0 + (unsigned)k;
        unsigned r     = (g << 1) + rowInG;
        int      idx   = idxs[r];
        unsigned table = r >> 17;
        uint64_t src = wB + (uint64_t)(((table * (unsigned)ROWS
                              + (unsigned)idx) << 8) + (chunk << 4)); // fits u32
        uint32_t ldsAddr = ldsBase + (uint32_t)(k * 512) + (lane << 4);
        asm volatile("global_load_async_to_lds_b128 %0, %1, off"
                     :
                     : "v"(ldsAddr), "v"(src)
                     : "memory");
    }

    // all LDS writes complete
    asm volatile("s_wait_asynccnt 0" ::: "memory");

    // ---- issue GPI async streaming stores LDS -> out ----
    #pragma unroll
    for (int k = 0; k < GPI; ++k) {
        unsigned g   = g0 + (unsigned)k;
        uint64_t dst = outB + (uint64_t)((g << 9) + (lane << 4)); // g*512+lane*16, fits u32
        uint32_t ldsAddr = ldsBase + (uint32_t)(k * 512) + (lane << 4);
        asm volatile("global_store_async_from_lds_b128 %0, %1, off"
                     :
                     : "v"(dst), "v"(ldsAddr)
                     : "memory");
    }

    asm volatile("s_wait_asynccnt 0" ::: "memory");
}

// ---------------------------------------------------------------------------
// Path A: direct gather, 16 threads per 256B row, one b128 each.
// All byte offsets fit in 32 bits (weights span 819MB, output 512MB), so we
// feed the backend zext-32-bit voffsets against a 64-bit SGPR base to get
// GVS addressing (global_load_b128 v, voff, s[base]) and minimal VALU work.
// Reads: default temporal hint (repeated rows stay in 192MB L2).
// Writes: non-temporal b128 (write-once stream, keep L2 for the gather).
// ---------------------------------------------------------------------------
__global__ __launch_bounds__(256) void gather_direct(
    const int* __restrict__ idxs, const float* __restrict__ w,
    float* __restrict__ out, unsigned rowBase, unsigned nRows)
{
    unsigned gid   = blockIdx.x * 256u + threadIdx.x;
    unsigned rloc  = gid >> 4;          // local row
    if (rloc >= nRows) return;
    unsigned r     = rowBase + rloc;    // global row in [0, 2^21)
    unsigned chunk = gid & 15u;         // which float4 of the row
    int      idx   = idxs[r];
    unsigned table = r >> 17;           // LOOKUPS == 2^17

    unsigned srcOff = ((table * (unsigned)ROWS + (unsigned)idx) << 8) + (chunk << 4);
    unsigned dstOff = (r << 8) + (chunk << 4);

    v4f v = *(const v4f*)((const char*)w + (size_t)srcOff);
    __builtin_nontemporal_store(v, (v4f*)((char*)out + (size_t)dstOff));
}

// ---------------------------------------------------------------------------
// Launcher: split rows 50/50 between the two equally-BW-bound paths.
//   direct: rows [0, 2^20)   -> 2^20 * 16 threads = 65536 blocks of 256
//   async : groups [2^19, 2^20) -> 524288 groups / 8 = 65536 waves = 8192 blocks
// ---------------------------------------------------------------------------
extern "C" void kernel_launch(void* const* d_in, const int* in_sizes, int n_in,
                              void* d_out, int out_size, void* d_ws, size_t ws_size,
                              hipStream_t stream) {
    const int*   indices = (const int*)d_in[0];   // [16, 131072] int32
    const float* weights = (const float*)d_in[1]; // [16, 200000, 64] f32
    float*       out     = (float*)d_out;         // [2^21, 64] f32
    (void)in_sizes; (void)n_in; (void)out_size; (void)d_ws; (void)ws_size;

    const unsigned halfRows   = TOTAL_ROWS / 2;            // 1,048,576
    const unsigned directBlks = (halfRows * 16u) / 256u;   // 65536
    gather_direct<<<directBlks, 256, 0, stream>>>(indices, weights, out, 0u, halfRows);

    const unsigned groupBase  = halfRows / 2;              // group g covers rows 2g,2g+1
    const unsigned nGroups    = halfRows / 2;              // 524,288
    const unsigned asyncBlks  = nGroups / (GPI * 8u);      // 8192 blocks (8 waves each)
    gather_async<<<asyncBlks, 256, 0, stream>>>(indices, weights, out, groupBase, nGroups);
}